// RTGNActorNet_72138270703872
// MI455X (gfx1250) — compile-verified
//
#include <hip/hip_runtime.h>
#include <math.h>

// ---------------------------------------------------------------------------
// RTGN actor net for gfx1250 (MI455X). fp32 WMMA (V_WMMA_F32_16X16X4_F32) for
// all node/edge GEMMs; message passing is a bandwidth-bound streaming kernel.
// ---------------------------------------------------------------------------

#define DIMD        64
#define N_NODES     5000
#define N_EDGES     20000
#define N_TORSIONS  1000
#define ACTION_DIM  6

typedef __attribute__((ext_vector_type(2))) float v2f;
typedef __attribute__((ext_vector_type(8))) float v8f;

__device__ __forceinline__ float sigm(float x) { return 1.0f / (1.0f + expf(-x)); }

// ---------------------------------------------------------------------------
// out = relu(x @ lin0_w + lin0_b)   [5000,3]@[3,64] -> tiny, flat kernel
// ---------------------------------------------------------------------------
__global__ void k_init_out(const float* __restrict__ x, const float* __restrict__ w,
                           const float* __restrict__ b, float* __restrict__ node) {
  int idx = blockIdx.x * blockDim.x + threadIdx.x;
  if (idx >= N_NODES * DIMD) return;
  int n = idx >> 6, c = idx & 63;
  float v = b[c];
  v = fmaf(x[n * 3 + 0], w[0 * DIMD + c], v);
  v = fmaf(x[n * 3 + 1], w[1 * DIMD + c], v);
  v = fmaf(x[n * 3 + 2], w[2 * DIMD + c], v);
  node[idx] = fmaxf(v, 0.0f);
}

__global__ void k_deg_count(const int* __restrict__ dst, float* __restrict__ deg) {
  int e = blockIdx.x * blockDim.x + threadIdx.x;
  if (e >= N_EDGES) return;
  atomicAdd(&deg[dst[e]], 1.0f);
}

__global__ void k_deg_inv(float* __restrict__ deg) {
  int n = blockIdx.x * blockDim.x + threadIdx.x;
  if (n >= N_NODES) return;
  deg[n] = 1.0f / fmaxf(deg[n], 1.0f);
}

// ---------------------------------------------------------------------------
// ew[e, :] = relu(edge_attr[e] * nn1_w + nn1_b) @ nn2_w + nn2_b
// [20000 x 64] @ [64 x 4096] -- the dominant GEMM (10.5 GFLOP), fp32 WMMA.
// One wave per 16x16 output tile; A (edge hidden) computed on the fly.
// ---------------------------------------------------------------------------
__global__ void k_ew_gemm(const float* __restrict__ edge_attr,
                          const float* __restrict__ nn1_w, const float* __restrict__ nn1_b,
                          const float* __restrict__ nn2_w, const float* __restrict__ nn2_b,
                          float* __restrict__ ew) {
  int wave = (blockIdx.x * blockDim.x + threadIdx.x) >> 5;
  int lane = threadIdx.x & 31;
  int et = wave >> 8;          // 1250 edge tiles
  int nt = wave & 255;         // 256 col tiles
  if (et >= N_EDGES / 16) return;
  int lo = lane & 15, hi = lane >> 4;
  int n0 = nt * 16;
  float ea = edge_attr[et * 16 + lo];       // EDGE_DIM == 1
  v8f c = {};
#pragma unroll
  for (int kk = 0; kk < DIMD; kk += 4) {
    int ka = kk + 2 * hi;                   // per ISA f32 A/B fragment layout
    v2f a, bb;
    a[0] = fmaxf(fmaf(ea, nn1_w[ka], nn1_b[ka]), 0.0f);
    a[1] = fmaxf(fmaf(ea, nn1_w[ka + 1], nn1_b[ka + 1]), 0.0f);
    bb[0] = nn2_w[(size_t)ka * 4096 + n0 + lo];
    bb[1] = nn2_w[(size_t)(ka + 1) * 4096 + n0 + lo];
    c = __builtin_amdgcn_wmma_f32_16x16x4_f32(false, a, false, bb, (short)0, c,
                                              false, false);
  }
  float bias = nn2_b[n0 + lo];
  size_t base = (size_t)(et * 16 + 8 * hi) * 4096 + n0 + lo;
#pragma unroll
  for (int r = 0; r < 8; ++r)
    ew[base + (size_t)r * 4096] = c[r] + bias;
}

// ---------------------------------------------------------------------------
// Generic fused GEMM: C = act(A[M,64] @ B[64,N] + bias + addend*rowscale)
// One wave per 16x16 tile; fp32 WMMA, K unrolled 16x. N multiple of 16.
// ---------------------------------------------------------------------------
__global__ void k_gemm64(const float* __restrict__ A, const float* __restrict__ B,
                         const float* __restrict__ bias,
                         const float* __restrict__ addend,
                         const float* __restrict__ rowscale,
                         float* __restrict__ C, int M, int N, int relu) {
  int wave = (blockIdx.x * blockDim.x + threadIdx.x) >> 5;
  int lane = threadIdx.x & 31;
  int ntiles = N >> 4;
  int mt = wave / ntiles;
  int nt = wave - mt * ntiles;
  if (mt * 16 >= M) return;                 // wave-uniform: EXEC stays all-ones
  int lo = lane & 15, hi = lane >> 4;
  int row = mt * 16 + lo;
  int arow = row < M ? row : M - 1;         // clamp A loads; stores are guarded
  int n0 = nt * 16;
  const float4* Arow = (const float4*)(A + (size_t)arow * DIMD);
  v8f c = {};
#pragma unroll
  for (int s = 0; s < 16; ++s) {
    float4 q = Arow[s];                     // k = 4s .. 4s+3
    v2f a, bb;
    a[0] = hi ? q.z : q.x;
    a[1] = hi ? q.w : q.y;
    int ka = 4 * s + 2 * hi;
    bb[0] = B[(size_t)ka * N + n0 + lo];
    bb[1] = B[(size_t)(ka + 1) * N + n0 + lo];
    c = __builtin_amdgcn_wmma_f32_16x16x4_f32(false, a, false, bb, (short)0, c,
                                              false, false);
  }
  float bv = bias ? bias[n0 + lo] : 0.0f;
#pragma unroll
  for (int r = 0; r < 8; ++r) {
    int rr = mt * 16 + 8 * hi + r;
    if (rr < M) {
      float v = c[r] + bv;
      if (addend) v = fmaf(addend[(size_t)rr * N + n0 + lo],
                           rowscale ? rowscale[rr] : 1.0f, v);
      if (relu) v = fmaxf(v, 0.0f);
      C[(size_t)rr * N + n0 + lo] = v;
    }
  }
}

// ---------------------------------------------------------------------------
// Per-edge message: msg[e] = node[src[e]] @ ew[e] (64x64), scatter-add to agg.
// One wave per edge; ew rows read as coalesced 128B lines (bandwidth bound).
// ---------------------------------------------------------------------------
__global__ void k_msg_scatter(const float* __restrict__ node,
                              const float* __restrict__ ew,
                              const int* __restrict__ src, const int* __restrict__ dst,
                              float* __restrict__ agg) {
  int wave = (blockIdx.x * blockDim.x + threadIdx.x) >> 5;
  int lane = threadIdx.x & 31;
  if (wave >= N_EDGES) return;
  int s = src[wave], d = dst[wave];
  const float* x = node + (size_t)s * DIMD;
  float x0 = x[lane], x1 = x[lane + 32];
  const float* W = ew + (size_t)wave * 4096;
  float m0 = 0.0f, m1 = 0.0f;
#pragma unroll 4
  for (int i = 0; i < DIMD; ++i) {
    float xi = (i < 32) ? __shfl(x0, i, 32) : __shfl(x1, i - 32, 32);
    m0 = fmaf(xi, W[i * DIMD + lane], m0);
    m1 = fmaf(xi, W[i * DIMD + lane + 32], m1);
  }
  atomicAdd(&agg[(size_t)d * DIMD + lane], m0);
  atomicAdd(&agg[(size_t)d * DIMD + lane + 32], m1);
}

// ---------------------------------------------------------------------------
// GRU gate combine: h = (1-z)*n + z*h  (in-place on node, which is h==out)
// ---------------------------------------------------------------------------
__global__ void k_gru_gates(const float* __restrict__ gi, const float* __restrict__ gh,
                            float* __restrict__ node) {
  int idx = blockIdx.x * blockDim.x + threadIdx.x;
  if (idx >= N_NODES * DIMD) return;
  int n = idx >> 6, c = idx & 63;
  const float* Gi = gi + (size_t)n * 192;
  const float* Gh = gh + (size_t)n * 192;
  float r = sigm(Gi[c] + Gh[c]);
  float z = sigm(Gi[64 + c] + Gh[64 + c]);
  float nn = tanhf(fmaf(r, Gh[128 + c], Gi[128 + c]));
  float h = node[idx];
  node[idx] = fmaf(1.0f - z, nn, z * h);
}

// ---------------------------------------------------------------------------
// Set2Set pooling, 6 iterations, single block of 256 threads.
// ---------------------------------------------------------------------------
__global__ void k_set2set(const float* __restrict__ node,
                          const float* __restrict__ wi, const float* __restrict__ wh,
                          const float* __restrict__ bi, const float* __restrict__ bh,
                          float* __restrict__ qout) {
  __shared__ float sh[64], sc[64], q[128], g[256], r[64], rbuf[256];
  int tid = threadIdx.x;
  if (tid < 64) { sh[tid] = 0.0f; sc[tid] = 0.0f; }
  if (tid < 128) q[tid] = 0.0f;
  __syncthreads();

  for (int it = 0; it < 6; ++it) {
    // LSTM gates: [1,128]@[128,256] + [1,64]@[64,256]
    float acc = bi[tid] + bh[tid];
    for (int k = 0; k < 128; ++k) acc = fmaf(q[k], wi[k * 256 + tid], acc);
    for (int k = 0; k < 64; ++k)  acc = fmaf(sh[k], wh[k * 256 + tid], acc);
    g[tid] = acc;
    __syncthreads();
    if (tid < 64) {
      float i = sigm(g[tid]), f = sigm(g[64 + tid]);
      float gg = tanhf(g[128 + tid]), o = sigm(g[192 + tid]);
      float cn = fmaf(f, sc[tid], i * gg);
      sc[tid] = cn;
      sh[tid] = o * tanhf(cn);
    }
    __syncthreads();

    // attention logits e_n = node[n] . sh, kept per-thread in registers
    float ev[20];
    int cnt = 0;
    float lmax = -INFINITY;
    for (int n = tid; n < N_NODES; n += 256) {
      const float* x = node + (size_t)n * DIMD;
      float e = 0.0f;
      for (int c = 0; c < DIMD; ++c) e = fmaf(x[c], sh[c], e);
      ev[cnt++] = e;
      lmax = fmaxf(lmax, e);
    }
    rbuf[tid] = lmax;
    __syncthreads();
    for (int s = 128; s > 0; s >>= 1) {
      if (tid < s) rbuf[tid] = fmaxf(rbuf[tid], rbuf[tid + s]);
      __syncthreads();
    }
    float mx = rbuf[0];
    __syncthreads();
    float lsum = 0.0f;
    for (int j = 0; j < cnt; ++j) lsum += expf(ev[j] - mx);
    rbuf[tid] = lsum;
    __syncthreads();
    for (int s = 128; s > 0; s >>= 1) {
      if (tid < s) rbuf[tid] += rbuf[tid + s];
      __syncthreads();
    }
    float invZ = 1.0f / rbuf[0];
    if (tid < 64) r[tid] = 0.0f;
    __syncthreads();
    // r = sum_n softmax(e)_n * node[n]
    cnt = 0;
    for (int n = tid; n < N_NODES; n += 256) {
      float w = expf(ev[cnt++] - mx) * invZ;
      const float* x = node + (size_t)n * DIMD;
      for (int c = 0; c < DIMD; ++c) atomicAdd(&r[c], w * x[c]);
    }
    __syncthreads();
    if (tid < 64) { q[tid] = sh[tid]; q[64 + tid] = r[tid]; }
    __syncthreads();
  }
  if (tid < 128) qout[tid] = q[tid];
}

// ---------------------------------------------------------------------------
// Memory LSTM cell on q_star; writes lstm_h/lstm_c to d_out tail + ws copy.
// ---------------------------------------------------------------------------
__global__ void k_memlstm(const float* __restrict__ q, const float* __restrict__ hx,
                          const float* __restrict__ cx,
                          const float* __restrict__ wi, const float* __restrict__ wh,
                          const float* __restrict__ bi, const float* __restrict__ bh,
                          float* __restrict__ hout, float* __restrict__ cout,
                          float* __restrict__ hws) {
  __shared__ float g[256];
  int tid = threadIdx.x;
  float acc = bi[tid] + bh[tid];
  for (int k = 0; k < 128; ++k) acc = fmaf(q[k], wi[k * 256 + tid], acc);
  for (int k = 0; k < 64; ++k)  acc = fmaf(hx[k], wh[k * 256 + tid], acc);
  g[tid] = acc;
  __syncthreads();
  if (tid < 64) {
    float i = sigm(g[tid]), f = sigm(g[64 + tid]);
    float gg = tanhf(g[128 + tid]), o = sigm(g[192 + tid]);
    float cn = fmaf(f, cx[tid], i * gg);
    float hn = o * tanhf(cn);
    hout[tid] = hn;
    cout[tid] = cn;
    hws[tid] = hn;
  }
}

// ---------------------------------------------------------------------------
// Head: z1 = relu([feat | rep] @ lin1_w + lin1_b), faithful to torch
// view(4*dim,-1).permute(1,0) and repeat_interleave index math.
// ---------------------------------------------------------------------------
__global__ void k_head1(const float* __restrict__ node, const int* __restrict__ nonring,
                        const float* __restrict__ lstm_h,
                        const float* __restrict__ w, const float* __restrict__ b,
                        float* __restrict__ z1) {
  int idx = blockIdx.x * blockDim.x + threadIdx.x;
  if (idx >= N_TORSIONS * DIMD) return;
  int t = idx >> 6, c = idx & 63;
  float acc = b[c];
  for (int j = 0; j < 4 * DIMD; ++j) {
    int f = j * N_TORSIONS + t;               // feat[t,j] = sel.flat[j*T + t]
    int row = nonring[f >> 6];
    float fv = node[(size_t)row * DIMD + (f & 63)];
    acc = fmaf(fv, w[j * DIMD + c], acc);
  }
  for (int j = 0; j < DIMD; ++j) {
    float rv = lstm_h[(t * DIMD + j) / N_TORSIONS];  // faithful repeat+view
    acc = fmaf(rv, w[(4 * DIMD + j) * DIMD + c], acc);
  }
  z1[idx] = fmaxf(acc, 0.0f);
}

__global__ void k_head2(const float* __restrict__ z1, const float* __restrict__ w,
                        const float* __restrict__ b, float* __restrict__ logits) {
  int idx = blockIdx.x * blockDim.x + threadIdx.x;
  if (idx >= N_TORSIONS * ACTION_DIM) return;
  int t = idx / ACTION_DIM, c = idx % ACTION_DIM;
  float acc = b[c];
  for (int k = 0; k < DIMD; ++k)
    acc = fmaf(z1[(size_t)t * DIMD + k], w[k * ACTION_DIM + c], acc);
  logits[idx] = acc;
}

// ---------------------------------------------------------------------------
// Launch
// ---------------------------------------------------------------------------
extern "C" void kernel_launch(void* const* d_in, const int* in_sizes, int n_in,
                              void* d_out, int out_size, void* d_ws, size_t ws_size,
                              hipStream_t stream) {
  const float* x         = (const float*)d_in[0];
  const int*   edge_idx  = (const int*)d_in[1];      // [2, E]
  const float* edge_attr = (const float*)d_in[2];
  // d_in[3] = batch (all zeros, single graph) -- unused
  const int*   nonring   = (const int*)d_in[4];
  const float* hx        = (const float*)d_in[5];
  const float* cx        = (const float*)d_in[6];
  const float* lin0_w    = (const float*)d_in[7];
  const float* lin0_b    = (const float*)d_in[8];
  const float* nn1_w     = (const float*)d_in[9];
  const float* nn1_b     = (const float*)d_in[10];
  const float* nn2_w     = (const float*)d_in[11];
  const float* nn2_b     = (const float*)d_in[12];
  const float* root_w    = (const float*)d_in[13];
  const float* conv_b    = (const float*)d_in[14];
  const float* gru_wih   = (const float*)d_in[15];
  const float* gru_whh   = (const float*)d_in[16];
  const float* gru_bih   = (const float*)d_in[17];
  const float* gru_bhh   = (const float*)d_in[18];
  const float* s2s_wi    = (const float*)d_in[19];
  const float* s2s_wh    = (const float*)d_in[20];
  const float* s2s_bi    = (const float*)d_in[21];
  const float* s2s_bh    = (const float*)d_in[22];
  const float* mem_wi    = (const float*)d_in[23];
  const float* mem_wh    = (const float*)d_in[24];
  const float* mem_bi    = (const float*)d_in[25];
  const float* mem_bh    = (const float*)d_in[26];
  const float* lin1_w    = (const float*)d_in[27];
  const float* lin1_b    = (const float*)d_in[28];
  const float* lin2_w    = (const float*)d_in[29];
  const float* lin2_b    = (const float*)d_in[30];

  const int* e_src = edge_idx;
  const int* e_dst = edge_idx + N_EDGES;

  // Workspace layout (floats). ew dominates: 20000*4096*4B = 327.7 MB;
  // total ~340 MB — assumes harness ws is sized for the reference's
  // materialized edge-weight tensor (same as the reference's [E,d,d]).
  float* ws   = (float*)d_ws;
  float* ew   = ws;
  float* node = ew + (size_t)N_EDGES * 4096;            // 320000
  float* mbuf = node + (size_t)N_NODES * DIMD;          // 320000
  float* gi   = mbuf + (size_t)N_NODES * DIMD;          // 960000
  float* gh   = gi + (size_t)N_NODES * 192;             // 960000
  float* agg  = gh + (size_t)N_NODES * 192;             // 320000
  float* deg  = agg + (size_t)N_NODES * DIMD;           // 5120 (padded)
  float* qst  = deg + 5120;                             // 128
  float* hws  = qst + 128;                              // 64
  float* z1   = hws + 64;                               // 64000

  float* out_logits = (float*)d_out;                    // [1000, 6]
  float* out_h = out_logits + N_TORSIONS * ACTION_DIM;  // [64]
  float* out_c = out_h + DIMD;                          // [64]

  // degree
  hipMemsetAsync(deg, 0, N_NODES * sizeof(float), stream);
  k_deg_count<<<(N_EDGES + 255) / 256, 256, 0, stream>>>(e_dst, deg);
  k_deg_inv<<<(N_NODES + 255) / 256, 256, 0, stream>>>(deg);

  // node init
  k_init_out<<<(N_NODES * DIMD + 255) / 256, 256, 0, stream>>>(x, lin0_w, lin0_b, node);

  // edge weight matrices (one-time big WMMA GEMM): 1250*256 waves, 8 waves/blk
  k_ew_gemm<<<(N_EDGES / 16) * 256 / 8, 256, 0, stream>>>(edge_attr, nn1_w, nn1_b,
                                                          nn2_w, nn2_b, ew);

  const int mtiles = (N_NODES + 15) / 16;  // 313
  for (int it = 0; it < 6; ++it) {
    hipMemsetAsync(agg, 0, (size_t)N_NODES * DIMD * sizeof(float), stream);
    k_msg_scatter<<<N_EDGES * 32 / 256, 256, 0, stream>>>(node, ew, e_src, e_dst, agg);
    // m = relu(agg/deg + node@root_w + conv_b)
    {
      int waves = mtiles * (DIMD / 16);
      k_gemm64<<<(waves * 32 + 255) / 256, 256, 0, stream>>>(
          node, root_w, conv_b, agg, deg, mbuf, N_NODES, DIMD, 1);
    }
    // gi = m @ gru_wih + bih ; gh = node @ gru_whh + bhh
    {
      int waves = mtiles * (192 / 16);
      k_gemm64<<<(waves * 32 + 255) / 256, 256, 0, stream>>>(
          mbuf, gru_wih, gru_bih, nullptr, nullptr, gi, N_NODES, 192, 0);
      k_gemm64<<<(waves * 32 + 255) / 256, 256, 0, stream>>>(
          node, gru_whh, gru_bhh, nullptr, nullptr, gh, N_NODES, 192, 0);
    }
    k_gru_gates<<<(N_NODES * DIMD + 255) / 256, 256, 0, stream>>>(gi, gh, node);
  }

  // Set2Set pooling -> q_star [128]
  k_set2set<<<1, 256, 0, stream>>>(node, s2s_wi, s2s_wh, s2s_bi, s2s_bh, qst);

  // memory LSTM -> lstm_h/lstm_c (also into d_out tail)
  k_memlstm<<<1, 256, 0, stream>>>(qst, hx, cx, mem_wi, mem_wh, mem_bi, mem_bh,
                                   out_h, out_c, hws);

  // head
  k_head1<<<(N_TORSIONS * DIMD + 255) / 256, 256, 0, stream>>>(
      node, nonring, hws, lin1_w, lin1_b, z1);
  k_head2<<<(N_TORSIONS * ACTION_DIM + 255) / 256, 256, 0, stream>>>(
      z1, lin2_w, lin2_b, out_logits);
}